// SE3TransformerJ_62483184222603
// MI455X (gfx1250) — compile-verified
//
#include <hip/hip_runtime.h>
#include <math.h>

#define NN 50000
#define NE 800000
#define EPSBN 1e-5f

typedef float v2f __attribute__((ext_vector_type(2)));
typedef float v8f __attribute__((ext_vector_type(8)));

struct KP {
  const float* node_feat;
  const float* edge_d;
  const float* edge_w;
  const int*   src;
  const int*   dst;
  const float* W1[6]; const float* W2[6]; const float* W3[6];
  const float* b1[6]; const float* b2[6]; const float* b3[6];
  const float* be1[6]; const float* be2[6]; const float* g1[6]; const float* g2[6];
  const float* self0L0; const float* self0L1; const float* self1L1;
  float* feat; float* u; float* R1;
  float* count; float* agg0; float* agg1; float* aggA; float* aggB;
  float* statsA; float* statsB; float* coef1; float* coef2;
  float* h0; float* h1;
  float* out;
};

__global__ void k_zero(float* p, int n) {
  int i = blockIdx.x * 256 + threadIdx.x;
  if (i < n) p[i] = 0.f;
}

// Per-edge prep: r, u, feat=[w,r]; count via atomics; 5 scalar sums for BN1 stats.
__global__ __launch_bounds__(256) void k_prep(KP P) {
  __shared__ float red[5];
  int tid = threadIdx.x;
  int e = blockIdx.x * 256 + tid;           // grid exactly covers NE
  if (tid < 5) red[tid] = 0.f;
  __syncthreads();
  float dx = P.edge_d[3*(size_t)e+0];
  float dy = P.edge_d[3*(size_t)e+1];
  float dz = P.edge_d[3*(size_t)e+2];
  float r  = sqrtf(dx*dx + dy*dy + dz*dz);
  float iv = 1.f / fmaxf(r, 1e-8f);
  P.u[3*(size_t)e+0] = dx*iv;
  P.u[3*(size_t)e+1] = dy*iv;
  P.u[3*(size_t)e+2] = dz*iv;
  float w = P.edge_w[e];
  P.feat[2*(size_t)e+0] = w;
  P.feat[2*(size_t)e+1] = r;
  atomicAdd(P.count + P.dst[e], 1.f);
  float vals[5] = { w, r, w*w, r*r, w*r };
  #pragma unroll
  for (int k = 0; k < 5; k++) {
    float v = vals[k];
    #pragma unroll
    for (int off = 1; off < 32; off <<= 1) v += __shfl_xor(v, off);
    if ((tid & 31) == 0) atomicAdd(&red[k], v);
  }
  __syncthreads();
  if (tid < 5) atomicAdd(P.statsA + tid, red[tid]);
}

// BN1 closed-form from 5 scalars -> folded coeffs: h1 = relu(w*cw + r*cr + cc)
__global__ void k_fin1(KP P) {
  int c = threadIdx.x;
  if (c >= 192) return;
  int net = c >> 5, cl = c & 31;
  float Ei = 1.f / (float)NE;
  float Ew = P.statsA[0]*Ei, Er = P.statsA[1]*Ei;
  float vw = P.statsA[2]*Ei - Ew*Ew;
  float vr = P.statsA[3]*Ei - Er*Er;
  float cv = P.statsA[4]*Ei - Ew*Er;
  float w0 = P.W1[net][cl], w1 = P.W1[net][32+cl];
  float b  = P.b1[net][cl];
  float mu  = w0*Ew + w1*Er + b;
  float var = w0*w0*vw + w1*w1*vr + 2.f*w0*w1*cv;
  float s = P.g1[net][cl] * rsqrtf(var + EPSBN);
  P.coef1[c]       = s*w0;
  P.coef1[192+c]   = s*w1;
  P.coef1[384+c]   = s*b + P.be1[net][cl] - s*mu;
}

// WMMA pass: h2pre = h1 @ W2 (per-net 32x32), accumulate per-channel sum/sumsq.
__global__ __launch_bounds__(256) void k_stats2(KP P) {
  int lane = threadIdx.x & 31, wv = threadIdx.x >> 5;
  int half = lane >> 4, nn = lane & 15, m = nn;
  int net = blockIdx.y;
  float cwv[16], crv[16], ccv[16];
  #pragma unroll
  for (int j = 0; j < 8; j++)
    #pragma unroll
    for (int s = 0; s < 2; s++) {
      int c = net*32 + 4*j + 2*half + s;
      cwv[2*j+s] = P.coef1[c];
      crv[2*j+s] = P.coef1[192+c];
      ccv[2*j+s] = P.coef1[384+c];
    }
  const float* W2p = P.W2[net];
  float w2f[32];
  #pragma unroll
  for (int t = 0; t < 2; t++)
    #pragma unroll
    for (int j = 0; j < 8; j++)
      #pragma unroll
      for (int v = 0; v < 2; v++)
        w2f[t*16 + 2*j + v] = W2p[(4*j + v + 2*half)*32 + 16*t + nn];
  float sum0 = 0.f, sum1 = 0.f, sq0 = 0.f, sq1 = 0.f;
  int gw = blockIdx.x*8 + wv;
  for (int it = 0; it < 10; ++it) {
    int e = (gw + 5000*it)*16 + m;
    float wv_ = P.feat[2*(size_t)e], rv = P.feat[2*(size_t)e+1];
    float h1v[16];
    #pragma unroll
    for (int q = 0; q < 16; q++)
      h1v[q] = fmaxf(0.f, fmaf(wv_, cwv[q], fmaf(rv, crv[q], ccv[q])));
    v8f a0 = {0,0,0,0,0,0,0,0}, a1 = {0,0,0,0,0,0,0,0};
    #pragma unroll
    for (int j = 0; j < 8; j++) {
      v2f av; av[0] = h1v[2*j]; av[1] = h1v[2*j+1];
      v2f b0; b0[0] = w2f[2*j]; b0[1] = w2f[2*j+1];
      v2f b1; b1[0] = w2f[16+2*j]; b1[1] = w2f[16+2*j+1];
      a0 = __builtin_amdgcn_wmma_f32_16x16x4_f32(false, av, false, b0, (short)0, a0, false, false);
      a1 = __builtin_amdgcn_wmma_f32_16x16x4_f32(false, av, false, b1, (short)0, a1, false, false);
    }
    #pragma unroll
    for (int r2 = 0; r2 < 8; r2++) {
      sum0 += a0[r2]; sq0 += a0[r2]*a0[r2];
      sum1 += a1[r2]; sq1 += a1[r2]*a1[r2];
    }
  }
  sum0 += __shfl_xor(sum0, 16); sq0 += __shfl_xor(sq0, 16);
  sum1 += __shfl_xor(sum1, 16); sq1 += __shfl_xor(sq1, 16);
  if (lane < 16) {
    int c0 = net*32 + nn, c1 = net*32 + 16 + nn;
    atomicAdd(P.statsB + 2*c0,     sum0);
    atomicAdd(P.statsB + 2*c0 + 1, sq0);
    atomicAdd(P.statsB + 2*c1,     sum1);
    atomicAdd(P.statsB + 2*c1 + 1, sq1);
  }
}

// BN2 folded coeffs: h2 = relu(A2*gemm + C2), C2 = be2 - s2*mean(gemm) (b2 cancels).
__global__ void k_fin2(KP P) {
  int c = threadIdx.x;
  if (c >= 192) return;
  int net = c >> 5, cl = c & 31;
  float Ei = 1.f / (float)NE;
  float mean = P.statsB[2*c] * Ei;
  float var  = P.statsB[2*c+1] * Ei - mean*mean;
  float s = P.g2[net][cl] * rsqrtf(var + EPSBN);
  P.coef2[c]     = s;
  P.coef2[192+c] = P.be2[net][cl] - s*mean;
}

// Main WMMA pass: h1 -> W2 WMMA -> BN2 -> (nets 0/1: W3 WMMA + layer0 messages+scatter;
//                                          nets 2-5: small W3 dots -> R1 per edge)
__global__ __launch_bounds__(256) void k_main(KP P) {
  __shared__ float lt[8][16*33];
  int lane = threadIdx.x & 31, wv = threadIdx.x >> 5;
  int half = lane >> 4, nn = lane & 15, m = nn;
  int net = blockIdx.y;
  float* tile = lt[wv];
  float cwv[16], crv[16], ccv[16];
  #pragma unroll
  for (int j = 0; j < 8; j++)
    #pragma unroll
    for (int s = 0; s < 2; s++) {
      int c = net*32 + 4*j + 2*half + s;
      cwv[2*j+s] = P.coef1[c];
      crv[2*j+s] = P.coef1[192+c];
      ccv[2*j+s] = P.coef1[384+c];
    }
  const float* W2p = P.W2[net];
  float w2f[32];
  #pragma unroll
  for (int t = 0; t < 2; t++)
    #pragma unroll
    for (int j = 0; j < 8; j++)
      #pragma unroll
      for (int v = 0; v < 2; v++)
        w2f[t*16 + 2*j + v] = W2p[(4*j + v + 2*half)*32 + 16*t + nn];
  float A2c0 = P.coef2[net*32+nn],    C2c0 = P.coef2[192+net*32+nn];
  float A2c1 = P.coef2[net*32+16+nn], C2c1 = P.coef2[192+net*32+16+nn];
  float w3f[32]; float b3v0 = 0.f, b3v1 = 0.f;
  float w3c[48]; float b3s[3] = {0.f, 0.f, 0.f};
  int outc = (net == 5) ? 3 : 1;
  if (net < 2) {
    const float* W3p = P.W3[net];
    #pragma unroll
    for (int t = 0; t < 2; t++)
      #pragma unroll
      for (int j = 0; j < 8; j++)
        #pragma unroll
        for (int v = 0; v < 2; v++)
          w3f[t*16 + 2*j + v] = W3p[(4*j + v + 2*half)*32 + 16*t + nn];
    b3v0 = P.b3[net][nn];
    b3v1 = P.b3[net][16+nn];
  } else {
    const float* W3p = P.W3[net];
    for (int o = 0; o < outc; o++) {
      for (int k = 0; k < 16; k++) w3c[o*16+k] = W3p[(16*half + k)*outc + o];
      b3s[o] = P.b3[net][o];
    }
  }
  int gw = blockIdx.x*8 + wv;
  for (int it = 0; it < 10; ++it) {
    int e = (gw + 5000*it)*16 + m;
    float wv_ = P.feat[2*(size_t)e], rv = P.feat[2*(size_t)e+1];
    float h1v[16];
    #pragma unroll
    for (int q = 0; q < 16; q++)
      h1v[q] = fmaxf(0.f, fmaf(wv_, cwv[q], fmaf(rv, crv[q], ccv[q])));
    v8f a0 = {0,0,0,0,0,0,0,0}, a1 = {0,0,0,0,0,0,0,0};
    #pragma unroll
    for (int j = 0; j < 8; j++) {
      v2f av; av[0] = h1v[2*j]; av[1] = h1v[2*j+1];
      v2f b0; b0[0] = w2f[2*j]; b0[1] = w2f[2*j+1];
      v2f b1; b1[0] = w2f[16+2*j]; b1[1] = w2f[16+2*j+1];
      a0 = __builtin_amdgcn_wmma_f32_16x16x4_f32(false, av, false, b0, (short)0, a0, false, false);
      a1 = __builtin_amdgcn_wmma_f32_16x16x4_f32(false, av, false, b1, (short)0, a1, false, false);
    }
    #pragma unroll
    for (int r2 = 0; r2 < 8; r2++) {
      int row = r2 + 8*half;
      tile[row*33 + nn]      = fmaxf(0.f, fmaf(A2c0, a0[r2], C2c0));
      tile[row*33 + 16 + nn] = fmaxf(0.f, fmaf(A2c1, a1[r2], C2c1));
    }
    asm volatile("s_wait_dscnt 0" ::: "memory");
    if (net < 2) {
      float a2v[16];
      #pragma unroll
      for (int j = 0; j < 8; j++)
        #pragma unroll
        for (int s = 0; s < 2; s++)
          a2v[2*j+s] = tile[m*33 + 4*j + 2*half + s];
      v8f r0 = {0,0,0,0,0,0,0,0}, r1 = {0,0,0,0,0,0,0,0};
      #pragma unroll
      for (int j = 0; j < 8; j++) {
        v2f av; av[0] = a2v[2*j]; av[1] = a2v[2*j+1];
        v2f b0; b0[0] = w3f[2*j]; b0[1] = w3f[2*j+1];
        v2f b1; b1[0] = w3f[16+2*j]; b1[1] = w3f[16+2*j+1];
        r0 = __builtin_amdgcn_wmma_f32_16x16x4_f32(false, av, false, b0, (short)0, r0, false, false);
        r1 = __builtin_amdgcn_wmma_f32_16x16x4_f32(false, av, false, b1, (short)0, r1, false, false);
      }
      #pragma unroll
      for (int r2 = 0; r2 < 8; r2++) {
        int row = r2 + 8*half;
        tile[row*33 + nn]      = r0[r2] + b3v0;
        tile[row*33 + 16 + nn] = r1[r2] + b3v1;
      }
      asm volatile("s_wait_dscnt 0" ::: "memory");
      int s = P.src[e], d = P.dst[e];
      const float4* nf4 = (const float4*)(P.node_feat + (size_t)s*32 + 16*half);
      float p = 0.f;
      #pragma unroll
      for (int k4 = 0; k4 < 4; k4++) {
        float4 f = nf4[k4];
        int cb = 16*half + 4*k4;
        p += f.x*tile[m*33+cb] + f.y*tile[m*33+cb+1]
           + f.z*tile[m*33+cb+2] + f.w*tile[m*33+cb+3];
      }
      p += __shfl_xor(p, 16);
      if (lane < 16) {
        if (net == 0) {
          atomicAdd(P.agg0 + d, p);
        } else {
          float ux = P.u[3*(size_t)e], uy = P.u[3*(size_t)e+1], uz = P.u[3*(size_t)e+2];
          atomicAdd(P.agg1 + 3*(size_t)d + 0, ux*p);
          atomicAdd(P.agg1 + 3*(size_t)d + 1, uy*p);
          atomicAdd(P.agg1 + 3*(size_t)d + 2, uz*p);
        }
      }
    } else {
      float hv[16];
      #pragma unroll
      for (int k = 0; k < 16; k++) hv[k] = tile[m*33 + 16*half + k];
      int sbase = (net == 5) ? 3 : (net - 2);
      for (int o = 0; o < outc; o++) {
        float p = 0.f;
        #pragma unroll
        for (int k = 0; k < 16; k++) p += hv[k]*w3c[o*16+k];
        p += __shfl_xor(p, 16);
        if (lane < 16) P.R1[(size_t)e*6 + sbase + o] = p + b3s[o];
      }
    }
  }
}

// Layer-0 node update: divide by count, add self-interaction (degree 0 only).
__global__ void k_node0(KP P) {
  int n = blockIdx.x*256 + threadIdx.x;
  if (n >= NN) return;
  float cnt = P.count[n];
  float inv = 1.f / fmaxf(cnt, 1.f);
  float hi = (cnt > 0.f) ? 1.f : 0.f;
  float dot = 0.f;
  for (int i = 0; i < 32; i++) dot += P.self0L0[i]*P.node_feat[(size_t)n*32+i];
  P.h0[n] = P.agg0[n]*inv + hi*dot;
  for (int p = 0; p < 3; p++) P.h1[3*(size_t)n+p] = P.agg1[3*(size_t)n+p]*inv;
}

// Layer-1 edge messages using stored radial outputs R1 and equivariant bases.
__global__ __launch_bounds__(256) void k_edge1(KP P) {
  int e = blockIdx.x*256 + threadIdx.x;     // grid exactly covers NE
  const float* R = P.R1 + (size_t)e*6;
  float R00 = R[0], R01 = R[1], R10 = R[2], Ra = R[3], Rb = R[4], Rc = R[5];
  float ux = P.u[3*(size_t)e], uy = P.u[3*(size_t)e+1], uz = P.u[3*(size_t)e+2];
  int s = P.src[e], d = P.dst[e];
  float h0s = P.h0[s];
  float hx = P.h1[3*(size_t)s], hy = P.h1[3*(size_t)s+1], hz = P.h1[3*(size_t)s+2];
  float udot = ux*hx + uy*hy + uz*hz;
  float msg0 = R00*h0s + R10*udot;
  const float i3 = 0.5773502691896258f;     // 1/sqrt(3)
  // cross = h1 x u
  float cx = hy*uz - hz*uy;
  float cy = hz*ux - hx*uz;
  float cz = hx*uy - hy*ux;
  float mx = R01*ux*h0s + Ra*hx*i3 + Rb*cx + Rc*(ux*udot - hx*(1.f/3.f));
  float my = R01*uy*h0s + Ra*hy*i3 + Rb*cy + Rc*(uy*udot - hy*(1.f/3.f));
  float mz = R01*uz*h0s + Ra*hz*i3 + Rb*cz + Rc*(uz*udot - hz*(1.f/3.f));
  atomicAdd(P.aggA + d, msg0);
  atomicAdd(P.aggB + 3*(size_t)d + 0, mx);
  atomicAdd(P.aggB + 3*(size_t)d + 1, my);
  atomicAdd(P.aggB + 3*(size_t)d + 2, mz);
}

// Layer-1 node update -> final outputs.
__global__ void k_node1(KP P) {
  int n = blockIdx.x*256 + threadIdx.x;
  if (n >= NN) return;
  float cnt = P.count[n];
  float inv = 1.f / fmaxf(cnt, 1.f);
  float hi = (cnt > 0.f) ? 1.f : 0.f;
  P.out[n] = P.aggA[n]*inv + hi*P.self0L1[0]*P.h0[n];
  for (int p = 0; p < 3; p++)
    P.out[NN + 3*(size_t)n + p] = P.aggB[3*(size_t)n+p]*inv + hi*P.self1L1[0]*P.h1[3*(size_t)n+p];
}

extern "C" void kernel_launch(void* const* d_in, const int* in_sizes, int n_in,
                              void* d_out, int out_size, void* d_ws, size_t ws_size,
                              hipStream_t stream) {
  (void)in_sizes; (void)n_in; (void)out_size; (void)ws_size;
  KP P;
  P.node_feat = (const float*)d_in[0];
  P.edge_d    = (const float*)d_in[1];
  P.edge_w    = (const float*)d_in[2];
  P.src       = (const int*)d_in[3];
  P.dst       = (const int*)d_in[4];
  // params pytree order: layers in list order, dict keys sorted:
  // radial dict: W1,W2,W3,b1,b2,b3,be1,be2,g1,g2 ; layer keys k00,k01,(k10,k11),self*
  const int base[6] = {5, 15, 26, 36, 46, 56};  // L0k00,L0k01,L1k00,L1k01,L1k10,L1k11
  for (int n = 0; n < 6; n++) {
    P.W1[n]  = (const float*)d_in[base[n]+0];
    P.W2[n]  = (const float*)d_in[base[n]+1];
    P.W3[n]  = (const float*)d_in[base[n]+2];
    P.b1[n]  = (const float*)d_in[base[n]+3];
    P.b2[n]  = (const float*)d_in[base[n]+4];
    P.b3[n]  = (const float*)d_in[base[n]+5];
    P.be1[n] = (const float*)d_in[base[n]+6];
    P.be2[n] = (const float*)d_in[base[n]+7];
    P.g1[n]  = (const float*)d_in[base[n]+8];
    P.g2[n]  = (const float*)d_in[base[n]+9];
  }
  P.self0L0 = (const float*)d_in[25];
  P.self0L1 = (const float*)d_in[66];
  P.self1L1 = (const float*)d_in[67];

  float* ws = (float*)d_ws;
  size_t o = 0;
  P.feat = ws + o; o += 2*(size_t)NE;
  P.u    = ws + o; o += 3*(size_t)NE;
  P.R1   = ws + o; o += 6*(size_t)NE;
  float* zbase = ws + o;
  P.count = ws + o; o += NN;
  P.agg0  = ws + o; o += NN;
  P.agg1  = ws + o; o += 3*(size_t)NN;
  P.aggA  = ws + o; o += NN;
  P.aggB  = ws + o; o += 3*(size_t)NN;
  P.statsA = ws + o; o += 8;
  P.statsB = ws + o; o += 384;
  size_t zlen = (size_t)((ws + o) - zbase);
  P.coef1 = ws + o; o += 576;
  P.coef2 = ws + o; o += 384;
  P.h0 = ws + o; o += NN;
  P.h1 = ws + o; o += 3*(size_t)NN;
  P.out = (float*)d_out;

  k_zero<<<(int)((zlen + 255)/256), 256, 0, stream>>>(zbase, (int)zlen);
  k_prep<<<NE/256, 256, 0, stream>>>(P);
  k_fin1<<<1, 192, 0, stream>>>(P);
  dim3 g(625, 6, 1);
  k_stats2<<<g, 256, 0, stream>>>(P);
  k_fin2<<<1, 192, 0, stream>>>(P);
  k_main<<<g, 256, 0, stream>>>(P);
  k_node0<<<(NN + 255)/256, 256, 0, stream>>>(P);
  k_edge1<<<NE/256, 256, 0, stream>>>(P);
  k_node1<<<(NN + 255)/256, 256, 0, stream>>>(P);
}